// QGHL_63367947485882
// MI455X (gfx1250) — compile-verified
//
#include <hip/hip_runtime.h>
#include <math.h>

#define NQV 8
#define DIMV 256
#define HALF_PI_F 1.5707963267948966f
#define XS_STRIDE 260   // 64-bank conflict-free for b64 A-fragment reads (260 % 64 == 4)

typedef float v2f __attribute__((ext_vector_type(2)));
typedef float v8f __attribute__((ext_vector_type(8)));

struct c32 { float x, y; };

__device__ __forceinline__ void dswait() {
    asm volatile("s_wait_dscnt 0" ::: "memory");
}
__device__ __forceinline__ c32 cadd(c32 a, c32 b) { return {a.x + b.x, a.y + b.y}; }
__device__ __forceinline__ c32 cmul(c32 a, c32 b) {
    return {a.x * b.x - a.y * b.y, a.x * b.y + a.y * b.x};
}

// ---- per-wave gate application on a 256-amplitude state in LDS ----
__device__ __forceinline__ void apply1(c32* s, int lane, int w,
                                       c32 u00, c32 u01, c32 u10, c32 u11) {
    int b = 7 - w;
#pragma unroll
    for (int p = 0; p < 4; ++p) {
        int pr = lane + 32 * p;                       // 128 pairs
        int i0 = ((pr >> b) << (b + 1)) | (pr & ((1 << b) - 1));
        int i1 = i0 | (1 << b);
        c32 a0 = s[i0], a1 = s[i1];
        s[i0] = cadd(cmul(u00, a0), cmul(u01, a1));
        s[i1] = cadd(cmul(u10, a0), cmul(u11, a1));
    }
    dswait();
}
__device__ __forceinline__ void rxg(c32* s, int lane, int w, float t) {
    float c = cosf(0.5f * t), sn = sinf(0.5f * t);
    apply1(s, lane, w, {c, 0.f}, {0.f, -sn}, {0.f, -sn}, {c, 0.f});
}
__device__ __forceinline__ void ryg(c32* s, int lane, int w, float t) {
    float c = cosf(0.5f * t), sn = sinf(0.5f * t);
    apply1(s, lane, w, {c, 0.f}, {-sn, 0.f}, {sn, 0.f}, {c, 0.f});
}
__device__ __forceinline__ void rzg(c32* s, int lane, int w, float t) {
    float c = cosf(0.5f * t), sn = sinf(0.5f * t);
    apply1(s, lane, w, {c, -sn}, {0.f, 0.f}, {0.f, 0.f}, {c, sn});
}
__device__ __forceinline__ int expand2(int g, int bl, int bh) {
    int t = ((g >> bl) << (bl + 1)) | (g & ((1 << bl) - 1));
    t = ((t >> bh) << (bh + 1)) | (t & ((1 << bh) - 1));
    return t;
}
__device__ __forceinline__ void cnotg(c32* s, int lane, int wc, int wt) {
    int b1 = 7 - wc, b2 = 7 - wt;
    int bl = b1 < b2 ? b1 : b2, bh = b1 < b2 ? b2 : b1;
#pragma unroll
    for (int p = 0; p < 2; ++p) {
        int t = expand2(lane + 32 * p, bl, bh);
        int i10 = t | (1 << b1);
        int i11 = i10 | (1 << b2);
        c32 a = s[i10], b = s[i11];
        s[i10] = b; s[i11] = a;
    }
    dswait();
}
__device__ __forceinline__ void swapg(c32* s, int lane, int wi, int wj) {
    int b1 = 7 - wi, b2 = 7 - wj;
    int bl = b1 < b2 ? b1 : b2, bh = b1 < b2 ? b2 : b1;
#pragma unroll
    for (int p = 0; p < 2; ++p) {
        int t = expand2(lane + 32 * p, bl, bh);
        int i01 = t | (1 << b2);
        int i10 = t | (1 << b1);
        c32 a = s[i01], b = s[i10];
        s[i01] = b; s[i10] = a;
    }
    dswait();
}
__device__ __forceinline__ void ixxg(c32* s, int lane, int wi, int wj, float th) {
    float c = cosf(0.5f * th), sn = sinf(0.5f * th);   // off-diag factor = -i*sn
    int b1 = 7 - wi, b2 = 7 - wj;
    int bl = b1 < b2 ? b1 : b2, bh = b1 < b2 ? b2 : b1;
#pragma unroll
    for (int p = 0; p < 2; ++p) {
        int t = expand2(lane + 32 * p, bl, bh);
        int i00 = t, i01 = t | (1 << b2), i10 = t | (1 << b1), i11 = i10 | (1 << b2);
        c32 a00 = s[i00], a01 = s[i01], a10 = s[i10], a11 = s[i11];
        s[i00] = {c * a00.x + sn * a11.y, c * a00.y - sn * a11.x};
        s[i11] = {c * a11.x + sn * a00.y, c * a11.y - sn * a00.x};
        s[i01] = {c * a01.x + sn * a10.y, c * a01.y - sn * a10.x};
        s[i10] = {c * a10.x + sn * a01.y, c * a10.y - sn * a01.x};
    }
    dswait();
}

// ---- Kernel 1: build the 256x256 circuit unitary, packed as WMMA B fragments.
// One column (basis state) per wave; 32 blocks x 8 waves = 256 columns.
__global__ __launch_bounds__(256) void qcnn_build_u(
        const float* __restrict__ wts, const float* __restrict__ cpl,
        float* __restrict__ bpr, float* __restrict__ bpi) {
    __shared__ c32 st[8 * 256];
    int lane = threadIdx.x & 31;
    int wv = threadIdx.x >> 5;
    int col = blockIdx.x * 8 + wv;
    c32* s = st + wv * 256;

    for (int k = lane; k < 256; k += 32) s[k] = {(k == col) ? 1.f : 0.f, 0.f};
    dswait();

    // initial per-qubit RX, RY, RZ
    for (int i = 0; i < NQV; ++i) {
        rxg(s, lane, i, wts[i]);
        ryg(s, lane, i, wts[i + NQV]);
        rzg(s, lane, i, wts[i + 2 * NQV]);
    }
    // all-pairs IsingXX
    for (int i = 0; i < NQV; ++i)
        for (int j = i + 1; j < NQV; ++j)
            ixxg(s, lane, i, j, cpl[i * NQV + j]);

    // edges from min-max-normalized coupling thresholded at 0.5
    float mn = cpl[0], mx = cpl[0];
    for (int k = 1; k < NQV * NQV; ++k) {
        float v = cpl[k];
        mn = fminf(mn, v); mx = fmaxf(mx, v);
    }
    float thr = 0.5f * (mn + mx);
    int widx = 3 * NQV;
    for (int i = 0; i < NQV; ++i)
        for (int j = i + 1; j < NQV; ++j) {
            if (!(cpl[i * NQV + j] > thr)) continue;
            int nonadj = (j - i) != 1;
            if (nonadj) swapg(s, lane, i, j);
            // conv
            rzg(s, lane, j, -HALF_PI_F);
            cnotg(s, lane, j, i);
            rzg(s, lane, i, wts[widx]);
            ryg(s, lane, j, wts[widx + 1]);
            cnotg(s, lane, i, j);
            ryg(s, lane, j, wts[widx + 2]);
            cnotg(s, lane, j, i);
            rzg(s, lane, i, HALF_PI_F);
            // pool
            rzg(s, lane, j, -HALF_PI_F);
            cnotg(s, lane, j, i);
            rzg(s, lane, i, wts[widx + 3]);
            ryg(s, lane, j, wts[widx + 4]);
            cnotg(s, lane, i, j);
            ryg(s, lane, j, wts[widx + 5]);
            widx += 6;
            if (nonadj) swapg(s, lane, i, j);
        }

    // st[n] = U[n][col]. B[d][n] = U[n][d], d = col.
    // Fragment (nt,kc): lane l, elem e holds B[kc*4 + 2*(l>>4) + e][nt*16 + (l&15)]
    int kc = col >> 2;
    int lhi = (col >> 1) & 1;
    int slot = col & 1;
    for (int n = lane; n < 256; n += 32) {
        c32 a = s[n];
        int nt = n >> 4;
        int ln = (lhi << 4) | (n & 15);
        int fi = ((nt * 64 + kc) * 32 + ln) * 2 + slot;
        bpr[fi] = a.x;
        bpi[fi] = a.y;
    }
}

// ---- Kernel 2: fused normalize + complex GEMM (f32 WMMA) + |.|^2 + PauliZ reduce.
// 64 batch rows / WG, 8 waves; wave w owns N-tiles {2w, 2w+1} x {re,im} x 4 M-tiles.
__global__ __launch_bounds__(256) void qcnn_gemm(
        const float* __restrict__ X, const float* __restrict__ bpr,
        const float* __restrict__ bpi, float* __restrict__ out) {
    __shared__ float xs[64 * XS_STRIDE];
    __shared__ float pn[256];
    __shared__ float invn[64];

    int tid = threadIdx.x;
    const float* Xg = X + (size_t)blockIdx.x * 64 * DIMV;

    // load 64x256 tile (float4, coalesced)
#pragma unroll
    for (int it = 0; it < 16; ++it) {
        int r = it * 4 + (tid >> 6);
        int c = (tid & 63) * 4;
        float4 v = *(const float4*)(Xg + r * DIMV + c);
        *(float4*)(&xs[r * XS_STRIDE + c]) = v;
    }
    __syncthreads();

    // row norms (4 threads per row)
    {
        int r = tid >> 2, seg = tid & 3;
        float ss = 0.f;
        for (int k = seg * 64; k < seg * 64 + 64; ++k) {
            float v = xs[r * XS_STRIDE + k];
            ss += v * v;
        }
        pn[tid] = ss;
    }
    __syncthreads();
    if (tid < 64)
        invn[tid] = 1.0f / (pn[4 * tid] + pn[4 * tid + 1] + pn[4 * tid + 2] + pn[4 * tid + 3]);

    int w = tid >> 5, lane = tid & 31;
    int m = lane & 15, hi = lane >> 4;
    int nt0 = 2 * w;
    const v2f* bpr2 = (const v2f*)bpr;
    const v2f* bpi2 = (const v2f*)bpi;

    v8f accr[4][2], acci[4][2];
#pragma unroll
    for (int mt = 0; mt < 4; ++mt)
#pragma unroll
        for (int nl = 0; nl < 2; ++nl) {
            accr[mt][nl] = (v8f)(0.f);
            acci[mt][nl] = (v8f)(0.f);
        }

#pragma unroll 2
    for (int kc = 0; kc < 64; ++kc) {
        int k0 = kc * 4 + hi * 2;
        v2f a[4];
#pragma unroll
        for (int mt = 0; mt < 4; ++mt)
            a[mt] = *(const v2f*)(&xs[(mt * 16 + m) * XS_STRIDE + k0]);
        v2f br[2], bi[2];
#pragma unroll
        for (int nl = 0; nl < 2; ++nl) {
            int fi = ((nt0 + nl) * 64 + kc) * 32 + lane;
            br[nl] = bpr2[fi];
            bi[nl] = bpi2[fi];
        }
#pragma unroll
        for (int mt = 0; mt < 4; ++mt)
#pragma unroll
            for (int nl = 0; nl < 2; ++nl) {
                accr[mt][nl] = __builtin_amdgcn_wmma_f32_16x16x4_f32(
                    false, a[mt], false, br[nl], (short)0, accr[mt][nl], false, false);
                acci[mt][nl] = __builtin_amdgcn_wmma_f32_16x16x4_f32(
                    false, a[mt], false, bi[nl], (short)0, acci[mt][nl], false, false);
            }
    }
    __syncthreads();   // all A-fragment reads done; reuse xs for probabilities

    // probs = yr^2 + yi^2 ; C layout: vgpr v -> M = v + 8*hi, N = m
#pragma unroll
    for (int mt = 0; mt < 4; ++mt)
#pragma unroll
        for (int nl = 0; nl < 2; ++nl) {
            v8f p = accr[mt][nl] * accr[mt][nl] + acci[mt][nl] * acci[mt][nl];
            int kk = (nt0 + nl) * 16 + m;
#pragma unroll
            for (int v = 0; v < 8; ++v)
                xs[(mt * 16 + v + 8 * hi) * XS_STRIDE + kk] = p[v];
        }
    __syncthreads();

    // out[b][q] = invn[b] * sum_k sign(q,k) * p[b][k], sign from bit (7-q) of k
    for (int o = tid; o < 512; o += 256) {
        int b = o >> 3, q = o & 7, sh = 7 - q;
        float sacc = 0.f;
        for (int k = 0; k < 256; ++k) {
            float p = xs[b * XS_STRIDE + k];
            sacc += ((k >> sh) & 1) ? -p : p;
        }
        out[((size_t)blockIdx.x * 64 + b) * 8 + q] = sacc * invn[b];
    }
}

extern "C" void kernel_launch(void* const* d_in, const int* in_sizes, int n_in,
                              void* d_out, int out_size, void* d_ws, size_t ws_size,
                              hipStream_t stream) {
    const float* x = (const float*)d_in[0];        // [B,256]
    const float* wts = (const float*)d_in[1];      // [192]
    const float* cpl = (const float*)d_in[2];      // [8,8]
    float* out = (float*)d_out;                    // [B,8]

    float* bpr = (float*)d_ws;                     // 16*64*32*2 floats = 256 KB
    float* bpi = bpr + 16 * 64 * 32 * 2;           // another 256 KB

    int B = in_sizes[0] / DIMV;                    // 65536

    qcnn_build_u<<<32, 256, 0, stream>>>(wts, cpl, bpr, bpi);
    qcnn_gemm<<<B / 64, 256, 0, stream>>>(x, bpr, bpi, out);
}